// lossx4_73967926772310
// MI455X (gfx1250) — compile-verified
//
#include <hip/hip_runtime.h>

typedef __attribute__((ext_vector_type(2))) float v2f;
typedef __attribute__((ext_vector_type(8))) float v8f;

constexpr int NB = 64;     // batch
constexpr int NC = 256;    // channels
constexpr int NH = 64;     // height == width
constexpr int NK = 21;     // keypoints
constexpr int PITCH = 65;  // LDS row pitch (floats) to avoid bank conflicts
constexpr int PLANE = NH * NH;

// ---------------------------------------------------------------------------
// Per-wave ROI pooling over one 64x64 plane resident in LDS.
// t[k,h] = sum_w mw[k,w]*F[h,w]  via V_WMMA_F32_16X16X4_F32 tiles,
// then rs[k] = sum_h mh[k,h]*t[k,h] via masked cross-lane reduction.
// bnd rows: 0=left(h lo) 1=right(h hi) 2=down(w lo) 3=up(w hi)
// ---------------------------------------------------------------------------
__device__ __forceinline__ void roi_wmma(const float* pl, const int (*bnd)[NK],
                                         float (*rsPart)[4], float* rout,
                                         int b, int c, int tid) {
  const int wid   = tid >> 5;
  const int lane  = tid & 31;
  const int lm    = lane & 15;
  const int half  = lane >> 4;
  const int mtile = (wid & 1) << 4;   // 0 or 16
  const int htile = (wid >> 1) << 4;  // 0,16,32,48
  const int m = mtile + lm;           // A-matrix row for this lane
  const int n = htile + lm;           // B/D column = h index

  // w-interval for this lane's A row (rows >= NK are zero-padded)
  int wlo = NH, whi = -1;
  if (m < NK) { wlo = bnd[2][m]; whi = bnd[3][m]; }

  v8f acc = {};
  const float* prow = pl + n * PITCH;
#pragma unroll
  for (int kk = 0; kk < 16; ++kk) {
    const int k0 = kk * 4 + half * 2;   // K indices {k0, k0+1} for this lane
    v2f a, bb;
    a.x = (k0     >= wlo && k0     < whi) ? 1.0f : 0.0f;
    a.y = (k0 + 1 >= wlo && k0 + 1 < whi) ? 1.0f : 0.0f;
    bb.x = prow[k0];
    bb.y = prow[k0 + 1];
    acc = __builtin_amdgcn_wmma_f32_16x16x4_f32(
        /*neg_a=*/false, a, /*neg_b=*/false, bb,
        /*c_mod=*/(short)0, acc, /*reuse_a=*/false, /*reuse_b=*/false);
  }

  // apply mh mask, reduce over the 16 h-columns held across lanes lm=0..15
#pragma unroll
  for (int r = 0; r < 8; ++r) {
    const int mr = mtile + half * 8 + r;   // D row -> keypoint index
    const int h  = htile + lm;             // D col -> h index
    float v = acc[r];
    float msk = 0.0f;
    if (mr < NK) {
      const int hlo = bnd[0][mr], hhi = bnd[1][mr];
      msk = (h >= hlo && h < hhi) ? 1.0f : 0.0f;
    }
    v *= msk;
    v += __shfl_xor(v, 1, 32);
    v += __shfl_xor(v, 2, 32);
    v += __shfl_xor(v, 4, 32);
    v += __shfl_xor(v, 8, 32);
    if (lm == 0) rsPart[mr][wid >> 1] = v;  // unique slot per (wave,half,r)
  }
  __syncthreads();
  if (tid < NK) {
    const float rs = rsPart[tid][0] + rsPart[tid][1] + rsPart[tid][2] + rsPart[tid][3];
    rout[((size_t)(b * NK + tid)) * NC + c] = rs;
  }
  __syncthreads();  // rsPart fully consumed before reuse
}

// ---------------------------------------------------------------------------
// K1: one block per (b,c) plane. Single HBM pass over f1,f2:
//   s[b,c] = sum(f1*f2);  r1/r2[b,k,c] = raw ROI sums (WMMA einsum)
// ---------------------------------------------------------------------------
__global__ __launch_bounds__(256) void plane_kernel(
    const float* __restrict__ f1, const float* __restrict__ f2,
    const float* __restrict__ pre1, const float* __restrict__ pre2,
    float* __restrict__ s_out, float* __restrict__ r1_out,
    float* __restrict__ r2_out) {
  __shared__ float pl1[NH * PITCH];
  __shared__ float pl2[NH * PITCH];
  __shared__ float red[256];
  __shared__ int bnd1[4][NK];
  __shared__ int bnd2[4][NK];
  __shared__ float rsPart[32][4];

  const int tid = threadIdx.x;
  const int blk = blockIdx.x;  // b*NC + c
  const int b = blk / NC;
  const int c = blk % NC;

  const float4* p1 = (const float4*)(f1 + (size_t)blk * PLANE);
  const float4* p2 = (const float4*)(f2 + (size_t)blk * PLANE);

  // stream both planes into LDS once; fuse the f1*f2 dot on the fly
  float dot = 0.0f;
#pragma unroll
  for (int j = 0; j < 4; ++j) {
    const int v = j * 256 + tid;          // float4 index within plane
    const float4 a = p1[v];
    const float4 q = p2[v];
    const int h = v >> 4;
    const int w = (v & 15) << 2;
    float* d1 = &pl1[h * PITCH + w];
    d1[0] = a.x; d1[1] = a.y; d1[2] = a.z; d1[3] = a.w;
    float* d2 = &pl2[h * PITCH + w];
    d2[0] = q.x; d2[1] = q.y; d2[2] = q.z; d2[3] = q.w;
    dot += a.x * q.x + a.y * q.y + a.z * q.z + a.w * q.w;
  }

  // interval bounds from pre (trunc-toward-zero of clipped coords, as torch .int())
  if (tid < NK) {
    const float x = pre1[(b * NK + tid) * 2 + 0];
    const float y = pre1[(b * NK + tid) * 2 + 1];
    bnd1[0][tid] = (int)fmaxf(x - 4.0f, 0.0f);
    bnd1[1][tid] = (int)fminf(x + 4.0f, 63.0f);
    bnd1[2][tid] = (int)fmaxf(y - 4.0f, 0.0f);
    bnd1[3][tid] = (int)fminf(y + 4.0f, 63.0f);
  } else if (tid >= 32 && tid < 32 + NK) {
    const int k = tid - 32;
    const float x = pre2[(b * NK + k) * 2 + 0];
    const float y = pre2[(b * NK + k) * 2 + 1];
    bnd2[0][k] = (int)fmaxf(x - 4.0f, 0.0f);
    bnd2[1][k] = (int)fminf(x + 4.0f, 63.0f);
    bnd2[2][k] = (int)fmaxf(y - 4.0f, 0.0f);
    bnd2[3][k] = (int)fminf(y + 4.0f, 63.0f);
  }

  red[tid] = dot;
  __syncthreads();
#pragma unroll
  for (int off = 128; off > 0; off >>= 1) {
    if (tid < off) red[tid] += red[tid + off];
    __syncthreads();
  }
  if (tid == 0) s_out[blk] = red[0];

  roi_wmma(pl1, bnd1, rsPart, r1_out, b, c, tid);
  roi_wmma(pl2, bnd2, rsPart, r2_out, b, c, tid);
}

// ---------------------------------------------------------------------------
// K2: smax = max(s) (m1,m2 cancel algebraically: w5 = s/max(s))
// ---------------------------------------------------------------------------
__global__ __launch_bounds__(256) void smax_kernel(const float* __restrict__ s,
                                                   float* __restrict__ smax) {
  __shared__ float red[256];
  const int tid = threadIdx.x;
  float m = -__builtin_huge_valf();
  for (int i = tid; i < NB * NC; i += 256) m = fmaxf(m, s[i]);
  red[tid] = m;
  __syncthreads();
#pragma unroll
  for (int off = 128; off > 0; off >>= 1) {
    if (tid < off) red[tid] = fmaxf(red[tid], red[tid + off]);
    __syncthreads();
  }
  if (tid == 0) smax[0] = red[0];
}

// ---------------------------------------------------------------------------
// K3: one block per keypoint; thread = channel.
//   fea = 0.999 * mean_b( (s[b,c]/smax) * rs[b,k,c] / cnt[b,k] ) + 0.001*val
//   partial[k] = sum_c (fea1-fea2)^2
// ---------------------------------------------------------------------------
__global__ __launch_bounds__(256) void combine_kernel(
    const float* __restrict__ s, const float* __restrict__ r1,
    const float* __restrict__ r2, const float* __restrict__ pre1,
    const float* __restrict__ pre2, const float* __restrict__ smax_p,
    const float* __restrict__ val1, const float* __restrict__ val2,
    float* __restrict__ partial) {
  __shared__ float cnt1[NB], cnt2[NB];
  __shared__ float red[256];
  const int k = blockIdx.x;
  const int c = threadIdx.x;

  if (c < NB) {  // thread index doubles as batch index for count precompute
    float x = pre1[(c * NK + k) * 2 + 0];
    float y = pre1[(c * NK + k) * 2 + 1];
    int l = (int)fmaxf(x - 4.0f, 0.0f), r = (int)fminf(x + 4.0f, 63.0f);
    int d = (int)fmaxf(y - 4.0f, 0.0f), u = (int)fminf(y + 4.0f, 63.0f);
    cnt1[c] = (float)((r - l) * (u - d));
    x = pre2[(c * NK + k) * 2 + 0];
    y = pre2[(c * NK + k) * 2 + 1];
    l = (int)fmaxf(x - 4.0f, 0.0f); r = (int)fminf(x + 4.0f, 63.0f);
    d = (int)fmaxf(y - 4.0f, 0.0f); u = (int)fminf(y + 4.0f, 63.0f);
    cnt2[c] = (float)((r - l) * (u - d));
  }
  __syncthreads();

  const float inv_smax = 1.0f / smax_p[0];
  float a1 = 0.0f, a2 = 0.0f;
  for (int b = 0; b < NB; ++b) {
    const float w5 = s[b * NC + c] * inv_smax;
    a1 += w5 * r1[((size_t)(b * NK + k)) * NC + c] / cnt1[b];
    a2 += w5 * r2[((size_t)(b * NK + k)) * NC + c] / cnt2[b];
  }
  const float fea1 = 0.999f * (a1 * (1.0f / NB)) + 0.001f * val1[0];
  const float fea2 = 0.999f * (a2 * (1.0f / NB)) + 0.001f * val2[0];
  const float diff = fea1 - fea2;
  red[c] = diff * diff;
  __syncthreads();
#pragma unroll
  for (int off = 128; off > 0; off >>= 1) {
    if (c < off) red[c] += red[c + off];
    __syncthreads();
  }
  if (c == 0) partial[k] = red[0];
}

__global__ void finish_kernel(const float* __restrict__ partial,
                              float* __restrict__ out) {
  if (threadIdx.x == 0) {
    float sum = 0.0f;
    for (int k = 0; k < NK; ++k) sum += partial[k];
    out[0] = sum / (float)(NK * NC);
  }
}

extern "C" void kernel_launch(void* const* d_in, const int* in_sizes, int n_in,
                              void* d_out, int out_size, void* d_ws,
                              size_t ws_size, hipStream_t stream) {
  const float* f1 = (const float*)d_in[0];
  const float* f2 = (const float*)d_in[1];
  const float* pre1 = (const float*)d_in[2];
  const float* pre2 = (const float*)d_in[3];
  const float* val1 = (const float*)d_in[4];
  const float* val2 = (const float*)d_in[5];

  float* ws = (float*)d_ws;
  float* s = ws;                          // 16384
  float* r1 = s + NB * NC;                // 344064
  float* r2 = r1 + NB * NK * NC;          // 344064
  float* smax = r2 + NB * NK * NC;        // 1
  float* partial = smax + 1;              // 21

  plane_kernel<<<NB * NC, 256, 0, stream>>>(f1, f2, pre1, pre2, s, r1, r2);
  smax_kernel<<<1, 256, 0, stream>>>(s, smax);
  combine_kernel<<<NK, 256, 0, stream>>>(s, r1, r2, pre1, pre2, smax, val1,
                                         val2, partial);
  finish_kernel<<<1, 32, 0, stream>>>(partial, (float*)d_out);
}